// LinearLayer_38860864094672
// MI455X (gfx1250) — compile-verified
//
#include <hip/hip_runtime.h>

typedef __attribute__((ext_vector_type(2))) float v2f;
typedef __attribute__((ext_vector_type(4))) float v4f;
typedef __attribute__((ext_vector_type(8))) float v8f;

#define EMB   128
#define BSZ   32
#define SEQ   200
#define TOUT  10
#define KITM  101
#define NTILE 7      // ceil(101 / 16)

// One wave (32 threads) per (b,t, item-tile).  16 gathered dot products of
// length 128 via 32 chained v_wmma_f32_16x16x4_f32 ops.
//
// K-permutation trick: the dot product is commutative over k, so we remap
// logical WMMA k-slots to memory indices such that each lane's data for TWO
// consecutive WMMAs is contiguous and fetched with one global_load_b128:
//   iteration j (0..15), lane-half h: memory elems [8j+4h .. 8j+4h+3]
//   -> .xy feeds wmma 2j, .zw feeds wmma 2j+1.
// A and B use the identical (j,h,vgpr)->memory mapping, so every product
// W[idx][i]*enc[i], i=0..127, is formed exactly once.
//
// A-fragment (16x4 f32, ISA layout): lane l (m=l&15, h=l>>4), VGPR v holds
//   A[m][kslot = v + 2h].  B-fragment (4x16): VGPR v, lane-half h holds
//   B[kslot = v + 2h][n] — we broadcast enc along N so value depends only on
//   the k-slot.  D (16x16 f32): lane l, VGPR r -> m = r + 8*(l>>4), n = l&15;
//   all N columns identical, lanes 0 and 16 (n==0) own the 16 results.
__global__ __launch_bounds__(32)
void gathered_linear_wmma(const float* __restrict__ enc,
                          const long long* __restrict__ items,
                          const float* __restrict__ W,
                          const float* __restrict__ bias,
                          float* __restrict__ out) {
  const int wid  = blockIdx.x;          // 0 .. B*T*NTILE-1
  const int tile = wid % NTILE;
  const int bt   = wid / NTILE;         // 0 .. B*T-1
  const int bi   = bt / TOUT;
  const int ti   = bt % TOUT;

  const int lane = threadIdx.x;         // 0..31 (wave32)
  const int half = lane >> 4;           // 0 or 1
  const int mrow = lane & 15;           // M row this lane feeds in A

  const float*     encRow  = enc + (size_t)(bi * SEQ + (SEQ - TOUT + ti)) * EMB;
  const long long* itemRow = items + (size_t)bt * KITM;

  // Gathered W row for this lane's A-matrix row (clamped for ragged tail;
  // results for padded rows are simply never stored).
  int pos  = tile * 16 + mrow;
  int posc = (pos < KITM) ? pos : (KITM - 1);
  int ridx = (int)itemRow[posc];
  const float* Wrow = W + (size_t)ridx * EMB;

  v8f acc = {0.f, 0.f, 0.f, 0.f, 0.f, 0.f, 0.f, 0.f};

#pragma unroll
  for (int j = 0; j < EMB / 8; ++j) {
    v4f a4 = *(const v4f*)(Wrow   + 8 * j + 4 * half);   // one b128 gather
    v4f b4 = *(const v4f*)(encRow + 8 * j + 4 * half);   // one b128 broadcast
    v2f a0 = {a4.x, a4.y}, a1 = {a4.z, a4.w};
    v2f b0 = {b4.x, b4.y}, b1 = {b4.z, b4.w};
    // 8 args: (neg_a, A, neg_b, B, c_mod, C, reuse_a, reuse_b)
    acc = __builtin_amdgcn_wmma_f32_16x16x4_f32(
        false, a0, false, b0, (short)0, acc, false, false);
    acc = __builtin_amdgcn_wmma_f32_16x16x4_f32(
        false, a1, false, b1, (short)0, acc, false, false);
  }

  // Lanes with n==0 (lanes 0 and 16) hold items [8*half .. 8*half+7] of the
  // tile in acc[0..7]; add gathered bias and store the valid ones.
  if (mrow == 0) {
    const int base = tile * 16 + 8 * half;
#pragma unroll
    for (int r = 0; r < 8; ++r) {
      int p = base + r;
      if (p < KITM) {
        int rr = (int)itemRow[p];
        out[(size_t)bt * KITM + p] = acc[r] + bias[rr];
      }
    }
  }
}

extern "C" void kernel_launch(void* const* d_in, const int* in_sizes, int n_in,
                              void* d_out, int out_size, void* d_ws, size_t ws_size,
                              hipStream_t stream) {
  (void)in_sizes; (void)n_in; (void)out_size; (void)d_ws; (void)ws_size;
  const float*     enc   = (const float*)d_in[0];     // (B, S, EMB) f32
  const long long* items = (const long long*)d_in[1]; // (B, T, K) int64
  const float*     W     = (const float*)d_in[2];     // (V+1, EMB) f32
  const float*     bias  = (const float*)d_in[3];     // (V+1,) f32
  float* out = (float*)d_out;                         // (B, T, K) f32

  dim3 grid(BSZ * TOUT * NTILE);   // 2240 single-wave workgroups
  dim3 block(32);
  gathered_linear_wmma<<<grid, block, 0, stream>>>(enc, items, W, bias, out);
}